// GCN_GNNModel_71768903516461
// MI455X (gfx1250) — compile-verified
//
#include <hip/hip_runtime.h>

#define N_NODES 102400
#define N_EDGES 1638400
#define C 128
#define SLICE (N_NODES * C)

typedef __attribute__((ext_vector_type(2))) float v2f;
typedef __attribute__((ext_vector_type(8))) float v8f;

// ---------------------------------------------------------------- utilities

__global__ void zero_int_kernel(int* __restrict__ p, int n) {
    int i = blockIdx.x * blockDim.x + threadIdx.x;
    if (i < n) p[i] = 0;
}

__global__ void count_deg_kernel(const int* __restrict__ ei, int* __restrict__ deg) {
    int e = blockIdx.x * blockDim.x + threadIdx.x;
    if (e < N_EDGES) atomicAdd(&deg[ei[N_EDGES + e]], 1);
}

// Single-block exclusive scan over N_NODES in-degrees -> rowptr (and cursor copy).
__global__ void scan_kernel(const int* __restrict__ deg, int* __restrict__ rowptr,
                            int* __restrict__ cursor) {
    __shared__ int buf[1024];
    __shared__ int carry;
    if (threadIdx.x == 0) carry = 0;
    __syncthreads();
    for (int base = 0; base < N_NODES; base += 1024) {
        int i = base + threadIdx.x;
        int v = (i < N_NODES) ? deg[i] : 0;
        buf[threadIdx.x] = v;
        __syncthreads();
        for (int off = 1; off < 1024; off <<= 1) {
            int t = (threadIdx.x >= off) ? buf[threadIdx.x - off] : 0;
            __syncthreads();
            buf[threadIdx.x] += t;
            __syncthreads();
        }
        int incl = buf[threadIdx.x];
        int c0 = carry;
        if (i < N_NODES) {
            int r = c0 + incl - v;   // exclusive
            rowptr[i] = r;
            cursor[i] = r;
        }
        __syncthreads();
        if (threadIdx.x == 1023) carry = c0 + incl;
        __syncthreads();
    }
    if (threadIdx.x == 0) rowptr[N_NODES] = carry;
}

__global__ void scatter_kernel(const int* __restrict__ ei, int* __restrict__ cursor,
                               int* __restrict__ csr_src) {
    int e = blockIdx.x * blockDim.x + threadIdx.x;
    if (e < N_EDGES) {
        int dst = ei[N_EDGES + e];
        int pos = atomicAdd(&cursor[dst], 1);
        csr_src[pos] = ei[e];
    }
}

// ---------------------------------------------------------------- WMMA GEMM

// One wave computes a 16x16 tile of A[rows,128] @ W[128,128] via v_wmma_f32_16x16x4_f32.
// A frag (16x4 f32, 2 VGPRs): lane<16 -> M=lane, K=k0..k0+1 ; lane>=16 -> M=lane-16, K=k0+2..k0+3
// B frag (4x16 f32, 2 VGPRs): VGPR v, lane l -> K = k0 + (l>>4)*2 + v, N = l&15
__device__ __forceinline__ v8f wmma_tile_f32(const float* __restrict__ A,
                                             const float* __restrict__ W,
                                             int row0, int col0, int lane, v8f acc) {
    const int m  = lane & 15;
    const int kk = (lane >> 4) << 1;                 // 0 or 2
    const float* ap = A + (size_t)(row0 + m) * C + kk;
    const float* bp = W + (size_t)kk * C + col0 + m;
#pragma unroll
    for (int k0 = 0; k0 < C; k0 += 4) {
        v2f a = *(const v2f*)(ap + k0);
        v2f b;
        b.x = bp[(size_t)k0 * C];
        b.y = bp[(size_t)(k0 + 1) * C];
        acc = __builtin_amdgcn_wmma_f32_16x16x4_f32(false, a, false, b,
                                                    (short)0, acc, false, false);
    }
    return acc;
}

// h = x @ W_g (no bias; bias added in the GCN aggregation kernel)
__global__ void gemm_h_kernel(const float* __restrict__ A, const float* __restrict__ W,
                              float* __restrict__ Out) {
    const int lane = threadIdx.x & 31;
    const int row0 = blockIdx.x << 4;
    const int col0 = (threadIdx.x >> 5) << 4;
    v8f acc = {0.f, 0.f, 0.f, 0.f, 0.f, 0.f, 0.f, 0.f};
    acc = wmma_tile_f32(A, W, row0, col0, lane, acc);
    const int col   = col0 + (lane & 15);
    const int rbase = row0 + ((lane >> 4) << 3);
#pragma unroll
    for (int v = 0; v < 8; ++v)
        Out[(size_t)(rbase + v) * C + col] = acc[v];
}

// out = relu(mean @ W_l + b_l + xprev @ W_r)   (fused dual GEMM)
__global__ void sage_gemm_kernel(const float* __restrict__ Mean, const float* __restrict__ Wl,
                                 const float* __restrict__ bl,   const float* __restrict__ Xp,
                                 const float* __restrict__ Wr,   float* __restrict__ Out) {
    const int lane = threadIdx.x & 31;
    const int row0 = blockIdx.x << 4;
    const int col0 = (threadIdx.x >> 5) << 4;
    v8f acc = {0.f, 0.f, 0.f, 0.f, 0.f, 0.f, 0.f, 0.f};
    acc = wmma_tile_f32(Mean, Wl, row0, col0, lane, acc);
    acc = wmma_tile_f32(Xp,   Wr, row0, col0, lane, acc);
    const int col   = col0 + (lane & 15);
    const int rbase = row0 + ((lane >> 4) << 3);
    const float bias = bl[col];
#pragma unroll
    for (int v = 0; v < 8; ++v)
        Out[(size_t)(rbase + v) * C + col] = fmaxf(acc[v] + bias, 0.f);
}

// ---------------------------------------------------------------- aggregation

// One wave per dst node; lane owns 4 channels (float4). CSR gather from L2, no fp atomics.
// out = relu( dinv_d * sum_s h[s]*dinv_s  +  h[d]*dinv_d^2  +  b )
__global__ void gcn_agg_kernel(const float* __restrict__ h, const int* __restrict__ rowptr,
                               const int* __restrict__ csr_src, const int* __restrict__ deg,
                               const float* __restrict__ bias, float* __restrict__ out) {
    const int node = (blockIdx.x << 3) + (threadIdx.x >> 5);
    const int lane = threadIdx.x & 31;
    const float4* h4 = (const float4*)h;
    const int beg = rowptr[node], end = rowptr[node + 1];
    float4 acc = make_float4(0.f, 0.f, 0.f, 0.f);
    for (int e = beg; e < end; ++e) {
        int s = csr_src[e];
        float ds = rsqrtf((float)deg[s] + 1.0f);
        float4 v = h4[(size_t)s * 32 + lane];
        acc.x += v.x * ds; acc.y += v.y * ds; acc.z += v.z * ds; acc.w += v.w * ds;
    }
    const float dd   = rsqrtf((float)deg[node] + 1.0f);
    const float self = dd * dd;
    float4 hv = h4[(size_t)node * 32 + lane];
    float4 b4 = ((const float4*)bias)[lane];
    float4 r;
    r.x = fmaxf(acc.x * dd + hv.x * self + b4.x, 0.f);
    r.y = fmaxf(acc.y * dd + hv.y * self + b4.y, 0.f);
    r.z = fmaxf(acc.z * dd + hv.z * self + b4.z, 0.f);
    r.w = fmaxf(acc.w * dd + hv.w * self + b4.w, 0.f);
    ((float4*)out)[(size_t)node * 32 + lane] = r;
}

// mean[d] = sum_s x[s] / max(deg,1)
__global__ void sage_agg_kernel(const float* __restrict__ x, const int* __restrict__ rowptr,
                                const int* __restrict__ csr_src, const int* __restrict__ deg,
                                float* __restrict__ mean) {
    const int node = (blockIdx.x << 3) + (threadIdx.x >> 5);
    const int lane = threadIdx.x & 31;
    const float4* x4 = (const float4*)x;
    const int beg = rowptr[node], end = rowptr[node + 1];
    float4 acc = make_float4(0.f, 0.f, 0.f, 0.f);
    for (int e = beg; e < end; ++e) {
        int s = csr_src[e];
        float4 v = x4[(size_t)s * 32 + lane];
        acc.x += v.x; acc.y += v.y; acc.z += v.z; acc.w += v.w;
    }
    const float inv = 1.0f / fmaxf((float)deg[node], 1.0f);
    float4 r = make_float4(acc.x * inv, acc.y * inv, acc.z * inv, acc.w * inv);
    ((float4*)mean)[(size_t)node * 32 + lane] = r;
}

// ---------------------------------------------------------------- launch

extern "C" void kernel_launch(void* const* d_in, const int* in_sizes, int n_in,
                              void* d_out, int out_size, void* d_ws, size_t ws_size,
                              hipStream_t stream) {
    (void)in_sizes; (void)n_in; (void)out_size; (void)ws_size;
    const float* x   = (const float*)d_in[0];
    const float* W_g = (const float*)d_in[1];
    const float* b_g = (const float*)d_in[2];
    const float* W_l = (const float*)d_in[3];
    const float* b_l = (const float*)d_in[4];
    const float* W_r = (const float*)d_in[5];
    const int*   ei  = (const int*)d_in[6];
    float* out = (float*)d_out;

    // workspace layout (256B-aligned offsets), total ~60.2 MB
    char* w = (char*)d_ws;
    int*   deg     = (int*)(w + 0);          //  N ints           (409600 B)
    int*   rowptr  = (int*)(w + 409600);     //  N+1 ints         (-> 409856 B padded)
    int*   cursor  = (int*)(w + 819456);     //  N ints           (409600 B)
    int*   csr_src = (int*)(w + 1229056);    //  E ints           (6553600 B)
    float* hbuf    = (float*)(w + 7782656);  //  N*C floats, reused as `mean`

    // Build CSR-by-dst (counting sort), once per call.
    zero_int_kernel<<<(N_NODES + 255) / 256, 256, 0, stream>>>(deg, N_NODES);
    count_deg_kernel<<<N_EDGES / 256, 256, 0, stream>>>(ei, deg);
    scan_kernel<<<1, 1024, 0, stream>>>(deg, rowptr, cursor);
    scatter_kernel<<<N_EDGES / 256, 256, 0, stream>>>(ei, cursor, csr_src);

    // Layer 0: GCN
    gemm_h_kernel<<<N_NODES / 16, 256, 0, stream>>>(x, W_g, hbuf);
    gcn_agg_kernel<<<N_NODES / 8, 256, 0, stream>>>(hbuf, rowptr, csr_src, deg, b_g, out);

    // Layers 1..3: SAGE (mean aggr)
    for (int i = 0; i < 3; ++i) {
        const float* xp = out + (size_t)i * SLICE;
        float*       o  = out + (size_t)(i + 1) * SLICE;
        sage_agg_kernel<<<N_NODES / 8, 256, 0, stream>>>(xp, rowptr, csr_src, deg, hbuf);
        sage_gemm_kernel<<<N_NODES / 16, 256, 0, stream>>>(hbuf, W_l + (size_t)i * C * C,
                                                           b_l + (size_t)i * C, xp,
                                                           W_r + (size_t)i * C * C, o);
    }
}